// MinibatchDiscrimination_84842783965676
// MI455X (gfx1250) — compile-verified
//
#include <hip/hip_runtime.h>

typedef __attribute__((ext_vector_type(2))) float v2f;
typedef __attribute__((ext_vector_type(8))) float v8f;

#define BATCH 256
#define KDIM  1024   // inner GEMM dimension; also row stride of M
#define OUTF  64
#define KERD  16

// ---------------------------------------------------------------------------
// Phase 1: M = x @ T   ([256 x 1024] = [256 x 1024] * [1024 x 1024])
// One wave32 computes one 16x16 f32 tile via V_WMMA_F32_16X16X4_F32,
// stepping K by 4. 8 waves per block, grid (8, 16) covers 64 x 16 tiles.
// ---------------------------------------------------------------------------
__global__ __launch_bounds__(256) void mbd_gemm_wmma(const float* __restrict__ x,
                                                     const float* __restrict__ T,
                                                     float* __restrict__ Mout) {
    const int lane = threadIdx.x & 31;
    const int wave = threadIdx.x >> 5;
    const int tileCol = (blockIdx.x * 8 + wave) * 16;  // 0..1008
    const int tileRow = blockIdx.y * 16;               // 0..240
    const int lo   = lane & 15;
    const int half = lane >> 4;   // 0 for lanes 0-15, 1 for lanes 16-31

    // A fragment (16x4 f32): lane lo holds row tileRow+lo; VGPR pair holds
    // K = k+2*half, k+2*half+1 -> one aligned float2 load per step.
    const float* aRow = x + (size_t)(tileRow + lo) * KDIM;
    // B fragment (4x16 f32): lane lo holds col tileCol+lo; VGPR pair holds
    // rows K = k+2*half, k+2*half+1 (T row stride = 1024).
    const float* bCol = T + tileCol + lo;

    v8f c = {};
    for (int k = 0; k < KDIM; k += 4) {
        const int ks = k + 2 * half;
        v2f a = *(const v2f*)(aRow + ks);           // 8B-aligned (ks even)
        v2f b;
        b.x = bCol[(size_t)ks * 1024];
        b.y = bCol[(size_t)(ks + 1) * 1024];
        // 8 args: (neg_a, A, neg_b, B, c_mod, C, reuse_a, reuse_b)
        c = __builtin_amdgcn_wmma_f32_16x16x4_f32(
                false, a, false, b, (short)0, c, false, false);
    }

    // C/D layout: VGPR r -> lanes 0-15: M=r, lanes 16-31: M=r+8; N = lane&15.
    float* out = Mout + (size_t)tileRow * KDIM + tileCol;
#pragma unroll
    for (int r = 0; r < 8; ++r) {
        const int row = r + 8 * half;
        out[(size_t)row * KDIM + lo] = c[r];
    }
}

// ---------------------------------------------------------------------------
// Phase 2: out[i,f] = sum_j exp(-sum_k |M[i,f,k]-M[j,f,k]|) - 1
// One block per i; 256 threads = 64 features x 4 j-slices; LDS reduction.
// M is L2-resident (1 MB), so plain b128 global loads are fine.
// ---------------------------------------------------------------------------
__global__ __launch_bounds__(256) void mbd_pairwise(const float* __restrict__ Mm,
                                                    float* __restrict__ out) {
    const int i  = blockIdx.x;
    const int f  = threadIdx.x & 63;
    const int js = threadIdx.x >> 6;   // 0..3

    // Keep M[i, f, 0..15] in registers (4 x float4 = b128 loads).
    const float4* mi4 = (const float4*)(Mm + (size_t)i * KDIM + f * KERD);
    float4 a0 = mi4[0], a1 = mi4[1], a2 = mi4[2], a3 = mi4[3];

    float acc = 0.0f;
    for (int j = js; j < BATCH; j += 4) {
        const float4* mj4 = (const float4*)(Mm + (size_t)j * KDIM + f * KERD);
        float4 b0 = mj4[0], b1 = mj4[1], b2 = mj4[2], b3 = mj4[3];
        float l1 = 0.0f;
        l1 += fabsf(a0.x - b0.x) + fabsf(a0.y - b0.y) + fabsf(a0.z - b0.z) + fabsf(a0.w - b0.w);
        l1 += fabsf(a1.x - b1.x) + fabsf(a1.y - b1.y) + fabsf(a1.z - b1.z) + fabsf(a1.w - b1.w);
        l1 += fabsf(a2.x - b2.x) + fabsf(a2.y - b2.y) + fabsf(a2.z - b2.z) + fabsf(a2.w - b2.w);
        l1 += fabsf(a3.x - b3.x) + fabsf(a3.y - b3.y) + fabsf(a3.z - b3.z) + fabsf(a3.w - b3.w);
        acc += __expf(-l1);   // j==i contributes exp(0)=1, cancelled by -1 below
    }

    __shared__ float part[4][OUTF];
    part[js][f] = acc;
    __syncthreads();
    if (js == 0) {
        float s = part[0][f] + part[1][f] + part[2][f] + part[3][f];
        out[(size_t)i * OUTF + f] = s - 1.0f;
    }
}

// ---------------------------------------------------------------------------
extern "C" void kernel_launch(void* const* d_in, const int* in_sizes, int n_in,
                              void* d_out, int out_size, void* d_ws, size_t ws_size,
                              hipStream_t stream) {
    const float* x = (const float*)d_in[0];   // [256, 1024]
    const float* T = (const float*)d_in[1];   // [1024, 1024]
    float* Mm = (float*)d_ws;                 // scratch: 256*1024 f32 = 1 MiB

    dim3 gemmGrid(8, 16);                     // 8 waves/block * 128 blocks = 1024 tiles
    mbd_gemm_wmma<<<gemmGrid, 256, 0, stream>>>(x, T, Mm);
    mbd_pairwise<<<BATCH, 256, 0, stream>>>(Mm, (float*)d_out);
}